// AttentionRNNDecoder_2911987827497
// MI455X (gfx1250) — compile-verified
//
#include <hip/hip_runtime.h>
#include <hip/hip_bf16.h>
#include <math.h>

typedef __attribute__((ext_vector_type(2))) float v2f;
typedef __attribute__((ext_vector_type(8))) float v8f;

#define EMBED 620
#define ECH   1000
#define DCH   1000
#define VOCAB 30000
#define SRC   50
#define TRG   50
#define BSZ   32

// ---------------------------------------------------------------------------
// WMMA fp32 GEMM, specialized M=32:  C[32,N] = A[32,K] @ W[N,K]^T (+bias)(+act)
// One wave computes a 32x16 column strip (two 16x16 accumulators sharing the
// B fragment, so weight traffic is fetched exactly once per N-tile).
// A-fragment lane layout (ISA 7.12.2, 32-bit A 16x4):
//   lanes 0-15 = rows M=0..15; VGPR0 holds K={k0, k0+2}, VGPR1 K={k0+1,k0+3}
// => per lane: contiguous pair A[row][k0+2*half .. +1].  Same pattern for the
// B fragment read out of row-major W (we multiply by W^T).
// Ragged N tiles: clamp the weight-row pointer and load unconditionally —
// D-column n depends only on B-column n, and OOB lanes skip their stores, so
// no zeroing/predication is needed (keeps EXEC untouched in the hot loop).
// NT=1 marks weight loads non-temporal (W_p streams 434MB/step and must not
// thrash the L2-resident W_c/W_ih/W_hh, which are reused across all 49 steps).
// ---------------------------------------------------------------------------
template <int ACT, int NT>
__global__ void k_gemm_m32(const float* __restrict__ A, int lda,
                           const float* __restrict__ W, int ldw,
                           const float* __restrict__ bias,
                           float* __restrict__ C, int ldc,
                           int N, int K)
{
    int wid  = (blockIdx.x * blockDim.x + threadIdx.x) >> 5;
    int lane = threadIdx.x & 31;
    int ntiles = (N + 15) >> 4;
    if (wid >= ntiles) return;              // wave-uniform exit: EXEC stays full
    int hl = lane >> 4;                     // lane half (0/1)
    int r  = lane & 15;
    int n  = wid * 16 + r;
    bool nok = (n < N);
    int nc  = nok ? n : (N - 1);            // clamped row, always valid
    const float* __restrict__ wrow = W + (size_t)nc * (size_t)ldw;
    const float* __restrict__ a0p  = A + (size_t)r        * (size_t)lda;
    const float* __restrict__ a1p  = A + (size_t)(r + 16) * (size_t)lda;

    v8f c0 = {}; v8f c1 = {};
    for (int k0 = 0; k0 < K; k0 += 4) {
        int k = k0 + (hl << 1);
        v2f a0; a0[0] = a0p[k]; a0[1] = a0p[k + 1];
        v2f a1; a1[0] = a1p[k]; a1[1] = a1p[k + 1];
        v2f bf;
        if (NT) { bf[0] = __builtin_nontemporal_load(wrow + k);
                  bf[1] = __builtin_nontemporal_load(wrow + k + 1); }
        else    { bf[0] = wrow[k]; bf[1] = wrow[k + 1]; }
        c0 = __builtin_amdgcn_wmma_f32_16x16x4_f32(false, a0, false, bf, (short)0, c0, false, false);
        c1 = __builtin_amdgcn_wmma_f32_16x16x4_f32(false, a1, false, bf, (short)0, c1, false, false);
    }
    if (!nok) return;
    float bv = bias ? bias[n] : 0.0f;
#pragma unroll
    for (int g = 0; g < 8; ++g) {
        int m = g + (hl << 3);              // C/D layout: VGPR g = rows g / g+8
        float v0 = c0[g] + bv;
        float v1 = c1[g] + bv;
        if (ACT == 1) { v0 = tanhf(v0); v1 = tanhf(v1); }
        C[(size_t)m        * ldc + n] = v0;
        C[(size_t)(m + 16) * ldc + n] = v1;
    }
}

// Generic WMMA fp32 GEMM: C[M,N] = A[M,K] @ W[N,K]^T + bias.  M % 16 == 0.
__global__ void k_gemm_wmma(const float* __restrict__ A, int lda,
                            const float* __restrict__ W, int ldw,
                            const float* __restrict__ bias,
                            float* __restrict__ C, int ldc,
                            int M, int N, int K)
{
    int wid  = (blockIdx.x * blockDim.x + threadIdx.x) >> 5;
    int lane = threadIdx.x & 31;
    int ntiles = (N + 15) >> 4;
    int mtiles = M >> 4;
    if (wid >= mtiles * ntiles) return;     // wave-uniform
    int mt = wid / ntiles;
    int nt = wid - mt * ntiles;
    int hl = lane >> 4;
    int r  = lane & 15;
    int n  = nt * 16 + r;
    bool nok = (n < N);
    int nc  = nok ? n : (N - 1);
    const float* __restrict__ wrow = W + (size_t)nc * (size_t)ldw;
    const float* __restrict__ ap   = A + (size_t)(mt * 16 + r) * (size_t)lda;

    v8f c = {};
    for (int k0 = 0; k0 < K; k0 += 4) {
        int k = k0 + (hl << 1);
        v2f a; a[0] = ap[k];   a[1] = ap[k + 1];
        v2f bf; bf[0] = wrow[k]; bf[1] = wrow[k + 1];
        c = __builtin_amdgcn_wmma_f32_16x16x4_f32(false, a, false, bf, (short)0, c, false, false);
    }
    if (!nok) return;
    float bv = bias ? bias[n] : 0.0f;
#pragma unroll
    for (int g = 0; g < 8; ++g) {
        int m = mt * 16 + g + (hl << 3);
        C[(size_t)m * ldc + n] = c[g] + bv;
    }
}

// embedded -> xcat[:,0:620] & comb2[:,0:620];  h_prev -> comb2[:,620:1620]
__global__ void k_embed_pack(const int* __restrict__ tok,
                             const float* __restrict__ emb,
                             const float* __restrict__ hprev,
                             float* __restrict__ xcat,
                             float* __restrict__ comb2)
{
    int idx = blockIdx.x * blockDim.x + threadIdx.x;
    const int width = EMBED + DCH;
    if (idx >= BSZ * width) return;
    int b = idx / width;
    int c = idx - b * width;
    if (c < EMBED) {
        float v = emb[(size_t)tok[b] * EMBED + c];
        xcat [b * (EMBED + 2 * ECH) + c] = v;
        comb2[b * (EMBED + DCH + 2 * ECH) + c] = v;
    } else {
        int j = c - EMBED;
        comb2[b * (EMBED + DCH + 2 * ECH) + EMBED + j] = hprev[b * DCH + j];
    }
}

// scores[s,b] = sum_j tanh(hproj[b,j] + annproj[s,b,j]) * w_a[j]
// softmax over b (axis=1, faithful to reference); one block per s.
__global__ void k_attn(const float* __restrict__ hproj,
                       const float* __restrict__ annproj,
                       const float* __restrict__ w_a,
                       float* __restrict__ attn_cur,
                       float* __restrict__ attns_out)
{
    __shared__ float red[256];
    __shared__ float sc[BSZ];
    int s = blockIdx.x;
    int tid = threadIdx.x;
    int b = tid >> 3;
    int g = tid & 7;
    const float* ap = annproj + ((size_t)s * BSZ + b) * DCH;
    const float* hp = hproj + b * DCH;
    float part = 0.0f;
    for (int j = g; j < DCH; j += 8)
        part += tanhf(hp[j] + ap[j]) * w_a[j];
    red[tid] = part;
    __syncthreads();
    if (g == 0) {
        float sum = 0.0f;
        for (int i = 0; i < 8; ++i) sum += red[tid + i];
        sc[b] = sum;
    }
    __syncthreads();
    if (tid < BSZ) {
        float mx = -1e30f;
        for (int i = 0; i < BSZ; ++i) mx = fmaxf(mx, sc[i]);
        float se = 0.0f;
        for (int i = 0; i < BSZ; ++i) se += expf(sc[i] - mx);
        float a = expf(sc[tid] - mx) / se;
        attn_cur [s * BSZ + tid] = a;
        attns_out[s * BSZ + tid] = a;
    }
}

// context[b,e] = sum_s attn[s,b]*ann[s,b,e]  -> xcat[:,620:] & comb2[:,1620:]
__global__ void k_context_pack(const float* __restrict__ attn,
                               const float* __restrict__ ann,
                               float* __restrict__ xcat,
                               float* __restrict__ comb2)
{
    int idx = blockIdx.x * blockDim.x + threadIdx.x;
    if (idx >= BSZ * 2 * ECH) return;
    int b = idx / (2 * ECH);
    int e = idx - b * 2 * ECH;
    float acc = 0.0f;
    for (int s = 0; s < SRC; ++s)
        acc += attn[s * BSZ + b] * ann[(size_t)s * BSZ * 2 * ECH + (size_t)b * 2 * ECH + e];
    xcat [b * (EMBED + 2 * ECH) + EMBED + e] = acc;
    comb2[b * (EMBED + DCH + 2 * ECH) + EMBED + DCH + e] = acc;
}

// GRU gate combine (gi/gh already include b_ih/b_hh from the GEMM bias).
__global__ void k_gru(const float* __restrict__ gi, const float* __restrict__ gh,
                      const float* __restrict__ hprev, float* __restrict__ hnext)
{
    int idx = blockIdx.x * blockDim.x + threadIdx.x;
    if (idx >= BSZ * DCH) return;
    int b = idx / DCH;
    int j = idx - b * DCH;
    const float* gib = gi + (size_t)b * 3 * DCH;
    const float* ghb = gh + (size_t)b * 3 * DCH;
    float r = 1.0f / (1.0f + expf(-(gib[j]           + ghb[j])));
    float z = 1.0f / (1.0f + expf(-(gib[DCH + j]     + ghb[DCH + j])));
    float n = tanhf(gib[2 * DCH + j] + r * ghb[2 * DCH + j]);
    hnext[idx] = (1.0f - z) * n + z * hprev[idx];
}

// log_softmax over VOCAB; one block per batch row.
__global__ void k_logsoftmax(const float* __restrict__ logits, float* __restrict__ outp)
{
    __shared__ float red[256];
    int b = blockIdx.x;
    int tid = threadIdx.x;
    const float* lg = logits + (size_t)b * VOCAB;
    float mx = -1e30f;
    for (int v = tid; v < VOCAB; v += 256) mx = fmaxf(mx, lg[v]);
    red[tid] = mx; __syncthreads();
    for (int off = 128; off; off >>= 1) {
        if (tid < off) red[tid] = fmaxf(red[tid], red[tid + off]);
        __syncthreads();
    }
    mx = red[0]; __syncthreads();
    float sm = 0.0f;
    for (int v = tid; v < VOCAB; v += 256) sm += expf(lg[v] - mx);
    red[tid] = sm; __syncthreads();
    for (int off = 128; off; off >>= 1) {
        if (tid < off) red[tid] += red[tid + off];
        __syncthreads();
    }
    float lse = mx + logf(red[0]);
    for (int v = tid; v < VOCAB; v += 256)
        outp[(size_t)b * VOCAB + v] = lg[v] - lse;
}

// ---------------------------------------------------------------------------
extern "C" void kernel_launch(void* const* d_in, const int* in_sizes, int n_in,
                              void* d_out, int out_size, void* d_ws, size_t ws_size,
                              hipStream_t stream)
{
    (void)in_sizes; (void)n_in; (void)out_size; (void)ws_size;
    const int*   tokens = (const int*)  d_in[0];
    const float* ann    = (const float*)d_in[1];
    const float* emb    = (const float*)d_in[2];
    const float* W_h    = (const float*)d_in[3];
    const float* b_h    = (const float*)d_in[4];
    const float* W_c    = (const float*)d_in[5];
    const float* b_c    = (const float*)d_in[6];
    const float* w_a    = (const float*)d_in[7];
    const float* W_ih   = (const float*)d_in[8];
    const float* W_hh   = (const float*)d_in[9];
    const float* b_ih   = (const float*)d_in[10];
    const float* b_hh   = (const float*)d_in[11];
    const float* W_p    = (const float*)d_in[12];
    const float* b_p    = (const float*)d_in[13];
    float* out = (float*)d_out;
    float* ws  = (float*)d_ws;

    // workspace layout (floats)
    float* annproj  = ws;                       // SRC*BSZ*DCH = 1,600,000
    float* hbuf0    = ws + 1600000;             // 32,000
    float* hbuf1    = ws + 1632000;             // 32,000
    float* hproj    = ws + 1664000;             // 32,000
    float* attn_cur = ws + 1696000;             // 1,600
    float* xcat     = ws + 1697600;             // 32*2620 = 83,840
    float* comb2    = ws + 1781440;             // 32*3620 = 115,840
    float* gi       = ws + 1897280;             // 32*3000 = 96,000
    float* gh       = ws + 1993280;             // 32*3000 = 96,000
    float* logits   = ws + 2089280;             // 32*30000 = 960,000
    float* hbuf[2]  = { hbuf0, hbuf1 };

    const size_t PRED = (size_t)BSZ * VOCAB;    // 960,000
    float* preds  = out;
    float* hidden = out + (size_t)TRG * PRED;   // 48,000,000
    float* attns  = hidden + (size_t)BSZ * DCH; // +32,000

    // preds[0] = 0, attns[0] = 0 (d_out is poisoned by the harness)
    hipMemsetAsync(preds, 0, PRED * sizeof(float), stream);
    hipMemsetAsync(attns, 0, (size_t)SRC * BSZ * sizeof(float), stream);

    // h0 = tanh(ann[0,:,ECH:] @ W_h^T + b_h)   (M=32, N=1000, K=1000)
    {
        int nt = (DCH + 15) / 16;
        k_gemm_m32<1, 0><<<(nt + 7) / 8, 256, 0, stream>>>(
            ann + ECH, 2 * ECH, W_h, ECH, b_h, hbuf[0], DCH, DCH, ECH);
    }
    // annproj[s*B+b,:] = ann[s,b,:] @ W_c[:,DCH:]^T + b_c   (M=1600, N=1000, K=2000)
    {
        int tiles = (SRC * BSZ / 16) * ((DCH + 15) / 16);
        k_gemm_wmma<<<(tiles + 7) / 8, 256, 0, stream>>>(
            ann, 2 * ECH, W_c + DCH, DCH + 2 * ECH, b_c, annproj, DCH,
            SRC * BSZ, DCH, 2 * ECH);
    }

    for (int t = 1; t < TRG; ++t) {
        const float* hprev = hbuf[(t - 1) & 1];
        float*       hnext = hbuf[t & 1];

        k_embed_pack<<<(BSZ * (EMBED + DCH) + 255) / 256, 256, 0, stream>>>(
            tokens + (t - 1) * BSZ, emb, hprev, xcat, comb2);

        // hproj = h @ W_c[:,:DCH]^T   (N=1000, K=1000)
        {
            int nt = (DCH + 15) / 16;
            k_gemm_m32<0, 0><<<(nt + 7) / 8, 256, 0, stream>>>(
                hprev, DCH, W_c, DCH + 2 * ECH, nullptr, hproj, DCH, DCH, DCH);
        }

        k_attn<<<SRC, 256, 0, stream>>>(hproj, annproj, w_a, attn_cur,
                                        attns + (size_t)t * SRC * BSZ);

        k_context_pack<<<(BSZ * 2 * ECH + 255) / 256, 256, 0, stream>>>(
            attn_cur, ann, xcat, comb2);

        // gi = [emb, ctx] @ W_ih^T + b_ih   (N=3000, K=2620)
        {
            int nt = (3 * DCH + 15) / 16;
            k_gemm_m32<0, 0><<<(nt + 7) / 8, 256, 0, stream>>>(
                xcat, EMBED + 2 * ECH, W_ih, EMBED + 2 * ECH, b_ih,
                gi, 3 * DCH, 3 * DCH, EMBED + 2 * ECH);
        }
        // gh = h @ W_hh^T + b_hh   (N=3000, K=1000)
        {
            int nt = (3 * DCH + 15) / 16;
            k_gemm_m32<0, 0><<<(nt + 7) / 8, 256, 0, stream>>>(
                hprev, DCH, W_hh, DCH, b_hh, gh, 3 * DCH, 3 * DCH, DCH);
        }

        k_gru<<<(BSZ * DCH + 255) / 256, 256, 0, stream>>>(gi, gh, hprev, hnext);

        // logits = [emb, h_prev, ctx] @ W_p^T + b_p   (N=30000, K=3620)
        // NT=1: W_p (434MB) streams from HBM each step; keep it out of the way
        // of the L2-resident recurrent weights.
        {
            int nt = VOCAB / 16;  // 1875, exact
            k_gemm_m32<0, 1><<<(nt + 7) / 8, 256, 0, stream>>>(
                comb2, EMBED + DCH + 2 * ECH, W_p, EMBED + DCH + 2 * ECH, b_p,
                logits, VOCAB, VOCAB, EMBED + DCH + 2 * ECH);
        }

        k_logsoftmax<<<BSZ, 256, 0, stream>>>(logits, preds + (size_t)t * PRED);
    }

    hipMemcpyAsync(hidden, hbuf[(TRG - 1) & 1], (size_t)BSZ * DCH * sizeof(float),
                   hipMemcpyDeviceToDevice, stream);
}